// LocalAggregation_55997783605451
// MI455X (gfx1250) — compile-verified
//
#include <hip/hip_runtime.h>
#include <stdint.h>

// Problem constants (from reference)
#define B_   4
#define N1_  4096
#define N2_  4096
#define C_   144
#define K_   32
#define RAD_ 0.2f
#define R2_  0.04f
#define BIG_ 1.0e30f
#define NT_  (N2_ / 16)   // 256 support tiles

typedef __attribute__((ext_vector_type(2))) float v2f;
typedef __attribute__((ext_vector_type(8))) float v8f;

// ---------------------------------------------------------------------------
// Kernel 1: fused pairwise-distance (WMMA f32 16x16x4) + masked top-K select.
// ONE WAVE PER WORKGROUP (barrier ops are hardware S_NOPs for single-wave
// groups, ISA 3.1) — each wave owns 32 queries.
//   d^2 tile = A(16x4) * B(4x16) + C  with
//   A = [-2qx, -2qy, -2qz, 1],  B = [sx; sy; sz; s2],  C = q^2 broadcast.
// Support tiles (48 xyz dwords + 16 mask dwords) are double-buffered into LDS
// with GLOBAL_LOAD_ASYNC_TO_LDS_B32 so the next tile's fetch overlaps the
// top-K insertion VALU of the current tile (ASYNCcnt pipeline).
// ---------------------------------------------------------------------------
__global__ __launch_bounds__(32) void topk_kernel(
    const float* __restrict__ qxyz,   // [B,N1,3]
    const float* __restrict__ sxyz,   // [B,N2,3]
    const float* __restrict__ smask,  // [B,N2]
    int*   __restrict__ outIdx,       // [B,N1,K]
    float* __restrict__ outMsk)       // [B,N1,K]
{
  const int lane  = threadIdx.x;
  const int b     = blockIdx.y;
  const int qbase = blockIdx.x * 32;

  __shared__ float sQ[4][32];     // qx,qy,qz,q2 for this wave's 32 queries
  __shared__ float sD[32][16];    // 32x16 d^2 tile (rows = queries)
  __shared__ float sS[2][64];     // double-buffered support tile:
                                  //   [0..47] xyz interleaved, [48..63] mask

  const float* xyzBase = sxyz + (size_t)b * N2_ * 3;
  const float* mskBase = smask + (size_t)b * N2_;

  // Async-stage one 16-point support tile into LDS buffer `buf`.
  auto issue_tile = [&](int t, int buf) {
#pragma unroll
    for (int i = 0; i < 2; ++i) {
      const int w = lane + i * 32;                     // dword 0..63
      const float* g = (w < 48) ? (xyzBase + (size_t)t * 48 + w)
                                : (mskBase + (size_t)t * 16 + (w - 48));
      const uint32_t ldsa = (uint32_t)(uintptr_t)&sS[buf][w];
      asm volatile("global_load_async_to_lds_b32 %0, %1, off"
                   :: "v"(ldsa), "v"((uint64_t)(uintptr_t)g)
                   : "memory");
    }
  };

  // Load this wave's 32 query points.
  {
    const float* qp = qxyz + ((size_t)b * N1_ + qbase + lane) * 3;
    float x = qp[0], y = qp[1], z = qp[2];
    sQ[0][lane] = x;
    sQ[1][lane] = y;
    sQ[2][lane] = z;
    sQ[3][lane] = x * x + y * y + z * z;
  }

  // Prologue: stage tile 0 while we set up WMMA operands.
  issue_tile(0, 0);

  __syncthreads();   // single-wave group: compiler fence + hw S_NOP

  const int  m  = lane & 15;
  const bool hi = lane >= 16;     // lanes 16-31 carry K=2,3 planes
  const int  hh = hi ? 8 : 0;

  // A operands for the two row-tiles (rows 0..15 / 16..31), per ISA 7.12.2:
  // lanes 0-15 hold K0/K1, lanes 16-31 hold K2/K3.
  v2f a0, a1;
  a0.x = hi ? (-2.0f * sQ[2][m])      : (-2.0f * sQ[0][m]);
  a0.y = hi ? 1.0f                     : (-2.0f * sQ[1][m]);
  a1.x = hi ? (-2.0f * sQ[2][16 + m]) : (-2.0f * sQ[0][16 + m]);
  a1.y = hi ? 1.0f                     : (-2.0f * sQ[1][16 + m]);

  // C init: q^2 broadcast per row (VGPR j = rows M=j / M=j+8).
  v8f c0, c1;
#pragma unroll
  for (int j = 0; j < 8; ++j) {
    c0[j] = sQ[3][hh + j];
    c1[j] = sQ[3][16 + hh + j];
  }

  // Per-lane top-32 sorted list (ascending d^2). bi<0 == empty slot.
  float bd[K_];
  int   bi[K_];
#pragma unroll
  for (int k = 0; k < K_; ++k) { bd[k] = BIG_; bi[k] = -1; }

  for (int t = 0; t < NT_; ++t) {
    const int cur = t & 1;

    // Prefetch tile t+1 into the other buffer, then retire tile t.
    if (t + 1 < NT_) {
      // WAR: prior DS reads of the buffer we are about to overwrite must
      // have completed before the async engine writes into it.
      asm volatile("s_wait_dscnt 0x0" ::: "memory");
      issue_tile(t + 1, cur ^ 1);
      // 4 async ops outstanding; <=2 left  =>  tile t's 2 ops retired.
      asm volatile("s_wait_asynccnt 0x2" ::: "memory");
    } else {
      asm volatile("s_wait_asynccnt 0x0" ::: "memory");
    }

    // Build B operand from the staged tile (column n = t*16 + m).
    const float sx = sS[cur][3 * m + 0];
    const float sy = sS[cur][3 * m + 1];
    const float sz = sS[cur][3 * m + 2];
    const float sm = sS[cur][48 + m];
    const float s2 = sx * sx + sy * sy + sz * sz + (sm > 0.0f ? 0.0f : BIG_);
    v2f bb;
    bb.x = hi ? sz : sx;
    bb.y = hi ? s2 : sy;

    // d^2 tiles: D = A*B + C.
    v8f d0 = __builtin_amdgcn_wmma_f32_16x16x4_f32(
        false, a0, false, bb, (short)0, c0, false, false);
    v8f d1 = __builtin_amdgcn_wmma_f32_16x16x4_f32(
        false, a1, false, bb, (short)0, c1, false, false);

    // Scatter tiles to LDS as [row 0..31][col 0..15].
#pragma unroll
    for (int j = 0; j < 8; ++j) {
      sD[hh + j][m]      = d0[j];
      sD[16 + hh + j][m] = d1[j];
    }
    __syncthreads();

    // Lane L selects over query row L: radius test + sorted insert.
#pragma unroll 1
    for (int cidx = 0; cidx < 16; ++cidx) {
      const float d = sD[lane][cidx];
      if (d <= R2_ && d < bd[K_ - 1]) {
        bd[K_ - 1] = d;
        bi[K_ - 1] = t * 16 + cidx;
#pragma unroll
        for (int s = K_ - 1; s > 0; --s) {
          const float dl = bd[s - 1], dh = bd[s];
          const int   il = bi[s - 1], ih = bi[s];
          const bool  sw = dh < dl;
          bd[s - 1] = sw ? dh : dl;
          bd[s]     = sw ? dl : dh;
          bi[s - 1] = sw ? ih : il;
          bi[s]     = sw ? il : ih;
        }
      }
    }
    __syncthreads();   // order selection reads before next tile's sD stores
  }

  // Emit indices (invalid slots padded with the nearest valid, per reference).
  const int q = qbase + lane;
  const size_t base = ((size_t)b * N1_ + q) * K_;
  const int firstI = (bi[0] >= 0) ? bi[0] : 0;
#pragma unroll
  for (int k = 0; k < K_; ++k) {
    const bool valid = bi[k] >= 0;
    outIdx[base + k] = valid ? bi[k] : firstI;
    outMsk[base + k] = valid ? 1.0f : 0.0f;
  }
}

// ---------------------------------------------------------------------------
// Kernel 2: gather neighbors, position-embed (rel-pos per 3-channel group),
// masked mean over K. One block per (n1, b); lanes 0-31 stage neighbor data
// in LDS, threads 0..143 reduce their channel. Features are L2-resident
// (9.4 MB << 192 MB L2), so the column-strided gather stays on-chip.
// ---------------------------------------------------------------------------
__global__ __launch_bounds__(160) void agg_kernel(
    const float* __restrict__ qxyz,    // [B,N1,3]
    const float* __restrict__ sxyz,    // [B,N2,3]
    const float* __restrict__ qmask,   // [B,N1]
    const float* __restrict__ feat,    // [B,C,N2]
    const int*   __restrict__ topkIdx, // [B,N1,K]
    const float* __restrict__ topkMsk, // [B,N1,K]
    float* __restrict__ outPre)        // [B,C,N1]
{
  const int n1 = blockIdx.x;
  const int b  = blockIdx.y;
  const int tid = threadIdx.x;

  __shared__ float relS[3][K_];
  __shared__ float wS[K_];
  __shared__ int   idS[K_];

  if (tid < K_) {
    const size_t base = ((size_t)b * N1_ + n1) * K_;
    const int   id = topkIdx[base + tid];
    const float mk = topkMsk[base + tid];
    const float qm = qmask[(size_t)b * N1_ + n1];
    const float* qp = qxyz + ((size_t)b * N1_ + n1) * 3;
    const float* sp = sxyz + ((size_t)b * N2_ + id) * 3;
    const float inv_r = 1.0f / RAD_;
    relS[0][tid] = (sp[0] - qp[0]) * inv_r;
    relS[1][tid] = (sp[1] - qp[1]) * inv_r;
    relS[2][tid] = (sp[2] - qp[2]) * inv_r;
    wS[tid]  = mk + (1.0f - qm);
    idS[tid] = id;
  }
  __syncthreads();

  if (tid < C_) {
    float wsum = 0.0f;
#pragma unroll
    for (int k = 0; k < K_; ++k) wsum += wS[k];

    const float* frow = feat + ((size_t)b * C_ + tid) * N2_;
    __builtin_prefetch(frow + idS[0], 0, 0);   // global_prefetch_b8

    const float* rel = relS[tid % 3];
    float acc = 0.0f;
#pragma unroll
    for (int k = 0; k < K_; ++k)
      acc += frow[idS[k]] * rel[k] * wS[k];

    const float o = (wsum > 0.0f) ? (acc / wsum) : 0.0f;
    outPre[((size_t)b * C_ + tid) * N1_ + n1] = o;
  }
}

// ---------------------------------------------------------------------------
// Kernel 3: training-mode BatchNorm1d over (B, N1) per channel + ReLU.
// One block (256 threads) per channel; two strided passes over 16384 values.
// ---------------------------------------------------------------------------
__global__ __launch_bounds__(256) void bn_relu_kernel(
    const float* __restrict__ pre,    // [B,C,N1]
    const float* __restrict__ gamma,  // [C]
    const float* __restrict__ beta,   // [C]
    float* __restrict__ out)          // [B,C,N1]
{
  const int c = blockIdx.x;
  const int tid = threadIdx.x;
  const int total = B_ * N1_;

  __shared__ float sSum[256];
  __shared__ float sSq[256];

  float s = 0.0f, sq = 0.0f;
  for (int e = tid; e < total; e += 256) {
    const int bb = e >> 12;        // / N1_
    const int n  = e & (N1_ - 1);  // % N1_
    const float x = pre[((size_t)bb * C_ + c) * N1_ + n];
    s += x; sq += x * x;
  }
  sSum[tid] = s; sSq[tid] = sq;
  __syncthreads();
  for (int off = 128; off > 0; off >>= 1) {
    if (tid < off) { sSum[tid] += sSum[tid + off]; sSq[tid] += sSq[tid + off]; }
    __syncthreads();
  }
  const float inv_n = 1.0f / (float)total;
  const float mean = sSum[0] * inv_n;
  const float var  = sSq[0] * inv_n - mean * mean;
  const float g = gamma[c], be = beta[c];
  const float scale = g * rsqrtf(var + 1e-5f);

  for (int e = tid; e < total; e += 256) {
    const int bb = e >> 12;
    const int n  = e & (N1_ - 1);
    const size_t off2 = ((size_t)bb * C_ + c) * N1_ + n;
    const float y = scale * (pre[off2] - mean) + be;
    out[off2] = fmaxf(y, 0.0f);
  }
}

// ---------------------------------------------------------------------------
extern "C" void kernel_launch(void* const* d_in, const int* in_sizes, int n_in,
                              void* d_out, int out_size, void* d_ws, size_t ws_size,
                              hipStream_t stream) {
  const float* qxyz  = (const float*)d_in[0];  // [B,N1,3]
  const float* sxyz  = (const float*)d_in[1];  // [B,N2,3]
  const float* qmask = (const float*)d_in[2];  // [B,N1]
  const float* smask = (const float*)d_in[3];  // [B,N2]
  const float* feat  = (const float*)d_in[4];  // [B,C,N2]
  const float* gamma = (const float*)d_in[5];  // [C]
  const float* beta  = (const float*)d_in[6];  // [C]
  float* out = (float*)d_out;                  // [B,C,N1]

  // Workspace layout
  char* ws = (char*)d_ws;
  const size_t idxBytes = (size_t)B_ * N1_ * K_ * sizeof(int);    // 2 MB
  const size_t mskBytes = (size_t)B_ * N1_ * K_ * sizeof(float);  // 2 MB
  int*   topkIdx = (int*)ws;
  float* topkMsk = (float*)(ws + idxBytes);
  float* pre     = (float*)(ws + idxBytes + mskBytes);            // 9.4 MB

  dim3 g1(N1_ / 32, B_);
  topk_kernel<<<g1, 32, 0, stream>>>(qxyz, sxyz, smask, topkIdx, topkMsk);

  dim3 g2(N1_, B_);
  agg_kernel<<<g2, 160, 0, stream>>>(qxyz, sxyz, qmask, feat,
                                     topkIdx, topkMsk, pre);

  bn_relu_kernel<<<C_, 256, 0, stream>>>(pre, gamma, beta, out);
}